// SWE_Pooling_661424963995
// MI455X (gfx1250) — compile-verified
//
#include <hip/hip_runtime.h>

// SWE pooling for MI455X (gfx1250, wave32).
// Pipeline: normalize W -> fp32 WMMA GEMM (S = W @ X^T, layout B,L,N) ->
// per-(b,l) LDS bitonic sort fused with uniform-grid interp + ref subtraction.
// Bottleneck is the 4096 x 8192-element sorts; GEMM (34 GFLOP) and HBM
// traffic (~0.5 GB @ 23.3 TB/s) are comparatively free, so fp32 WMMA
// (V_WMMA_F32_16X16X4_F32) is used to keep sort keys bit-faithful.

typedef float v2f __attribute__((ext_vector_type(2)));
typedef float v8f __attribute__((ext_vector_type(8)));

constexpr int Bv = 16;
constexpr int Nv = 8192;
constexpr int Dv = 512;
constexpr int Lv = 256;
constexpr int Mv = 128;

// ---------------- Kernel 1: W = theta_v / ||row||_2 ----------------
__global__ __launch_bounds__(256)
void k_normalize_rows(const float* __restrict__ theta, float* __restrict__ W) {
    __shared__ float red[256];
    const int l = blockIdx.x;
    const int t = threadIdx.x;
    float ss = 0.f;
    for (int k = t; k < Dv; k += 256) {
        const float v = theta[l * Dv + k];
        ss += v * v;
    }
    red[t] = ss;
    __syncthreads();
    for (int off = 128; off > 0; off >>= 1) {
        if (t < off) red[t] += red[t + off];
        __syncthreads();
    }
    const float inv = 1.0f / sqrtf(red[0]);
    for (int k = t; k < Dv; k += 256) W[l * Dv + k] = theta[l * Dv + k] * inv;
}

// ---------------- Kernel 2: S[b][l][n] = dot(X[b,n,:], W[l,:]) via WMMA ----
// Block = 256 threads (8 wave32). Block computes a 16(l) x 128(n) strip.
// Wave w owns the 16x16 tile at n = nb + 16*w. K swept in chunks of 32.
__global__ __launch_bounds__(256)
void k_project_wmma(const float* __restrict__ X, const float* __restrict__ W,
                    float* __restrict__ S) {
    __shared__ float Wl[16][516];   // padded stride 516: banks 4*lane -> conflict-free
    __shared__ float Xc[128][33];   // padded stride 33: banks 33*lane -> conflict-free

    const int t  = threadIdx.x;
    const int b  = blockIdx.z;
    const int lt = blockIdx.y * 16;
    const int nb = blockIdx.x * 128;

    // Stage 16 full W rows (16 x 512 fp32) once.
    for (int idx = t; idx < 16 * 512; idx += 256) {
        const int r = idx >> 9, c = idx & 511;
        Wl[r][c] = W[(lt + r) * Dv + c];
    }

    const int wave = t >> 5;
    const int lane = t & 31;
    const int lm   = lane & 15;   // M (for A) / N (for B) within tile
    const int hi   = lane >> 4;   // half-wave selects K pair {0,1} vs {2,3}
    const int nw   = wave * 16;

    const float* Xb = X + (size_t)b * Nv * Dv + (size_t)nb * Dv;

    v8f acc = {};
    for (int kc = 0; kc < Dv; kc += 32) {
        __syncthreads();
        // Stage X chunk: 128 rows x 32 k, coalesced float4 loads.
        for (int idx = t; idx < 128 * 8; idx += 256) {
            const int n  = idx >> 3;
            const int k4 = (idx & 7) << 2;
            const float4 v = *(const float4*)(Xb + (size_t)n * Dv + kc + k4);
            Xc[n][k4 + 0] = v.x; Xc[n][k4 + 1] = v.y;
            Xc[n][k4 + 2] = v.z; Xc[n][k4 + 3] = v.w;
        }
        __syncthreads();

#pragma unroll
        for (int kk = 0; kk < 32; kk += 4) {
            // A (16x4 f32): lanes 0-15 hold K={0,1}, lanes 16-31 hold K={2,3}.
            const int ka = kc + kk + 2 * hi;
            v2f a; a.x = Wl[lm][ka]; a.y = Wl[lm][ka + 1];
            // B (4x16 f32): mirrored layout, N striped across lanes.
            const int kb = kk + 2 * hi;
            v2f bv; bv.x = Xc[nw + lm][kb]; bv.y = Xc[nw + lm][kb + 1];
            acc = __builtin_amdgcn_wmma_f32_16x16x4_f32(
                false, a, false, bv, (short)0, acc, false, false);
        }
    }

    // C/D layout: VGPR j -> row j (lanes 0-15) / row j+8 (lanes 16-31), col = lane%16.
    float* Sp = S + ((size_t)b * Lv + lt) * Nv;
#pragma unroll
    for (int j = 0; j < 8; ++j) {
        Sp[(size_t)(j + 8 * hi) * Nv + nb + nw + lm] = acc[j];
    }
}

// ---------------- Kernel 3: Rsorted[l][m] = sort_m( ref @ W[l] ) ----------
__global__ __launch_bounds__(128)
void k_ref_sorted(const float* __restrict__ refp, const float* __restrict__ W,
                  float* __restrict__ Rs) {
    __shared__ float wl[512];
    __shared__ float r[128];
    const int l = blockIdx.x;
    const int t = threadIdx.x;   // 128 threads, one per reference point
    for (int k = t; k < Dv; k += 128) wl[k] = W[l * Dv + k];
    __syncthreads();
    float s = 0.f;
    for (int k = 0; k < Dv; ++k) s += refp[t * Dv + k] * wl[k];
    r[t] = s;
    for (int size = 2; size <= 128; size <<= 1) {
        for (int stride = size >> 1; stride > 0; stride >>= 1) {
            __syncthreads();
            if (t < 64) {
                const int low = t & (stride - 1);
                const int i   = ((t & ~(stride - 1)) << 1) | low;
                const int j   = i | stride;
                const bool up = ((i & size) == 0);
                const float a = r[i], c = r[j];
                if ((a > c) == up) { r[i] = c; r[j] = a; }
            }
        }
    }
    __syncthreads();
    Rs[l * Mv + t] = r[t];
}

// ---------------- Kernel 4: per-(b,l) bitonic sort in LDS + interp + emb ---
__global__ __launch_bounds__(256)
void k_sort_interp_emb(const float* __restrict__ S, const float* __restrict__ Rs,
                       float* __restrict__ out) {
    __shared__ float s[Nv];   // 32 KB of the 320 KB WGP LDS
    const int l = blockIdx.x;
    const int b = blockIdx.y;
    const int t = threadIdx.x;

    const float* row = S + ((size_t)b * Lv + l) * Nv;
    for (int i = t; i < Nv; i += 256) s[i] = row[i];

    // Full bitonic network: 91 passes, 16 compare-exchanges per thread each.
    for (int size = 2; size <= Nv; size <<= 1) {
        for (int stride = size >> 1; stride > 0; stride >>= 1) {
            __syncthreads();
            for (int p = t; p < Nv / 2; p += 256) {
                const int low = p & (stride - 1);
                const int i   = ((p & ~(stride - 1)) << 1) | low;
                const int j   = i | stride;
                const bool up = ((i & size) == 0);
                const float a = s[i], c = s[j];
                if ((a > c) == up) { s[i] = c; s[j] = a; }
            }
        }
    }
    __syncthreads();

    if (t < Mv) {
        const int m = t;
        // Uniform grids: closed-form searchsorted.
        const long num  = (long)(m + 1) * (Nv + 1);
        const int idx0  = (int)((num - 1) / (Mv + 1));
        int ind = idx0 - 1;
        if (ind < 0) ind = 0;
        if (ind > Nv - 2) ind = Nv - 2;

        const float invX = 1.0f / (float)(Nv + 1);
        const float invM = 1.0f / (float)(Mv + 1);
        const float xi   = (float)(ind + 1) * invX;
        const float dx   = (float)(ind + 2) * invX - xi;
        const float xm   = (float)(m + 1) * invM;
        const float eps  = 1.1920929e-7f;

        const float slope = (s[ind + 1] - s[ind]) / (eps + dx);
        const float val   = s[ind] + slope * (xm - xi);
        out[(size_t)b * (Mv * Lv) + (size_t)m * Lv + l] = val - Rs[l * Mv + m];
    }
}

extern "C" void kernel_launch(void* const* d_in, const int* in_sizes, int n_in,
                              void* d_out, int out_size, void* d_ws, size_t ws_size,
                              hipStream_t stream) {
    (void)in_sizes; (void)n_in; (void)out_size; (void)ws_size;

    const float* X     = (const float*)d_in[0];  // (B, N, D)
    const float* theta = (const float*)d_in[1];  // (L, D)
    const float* refp  = (const float*)d_in[2];  // (M, D)
    float* out = (float*)d_out;                  // (B, M*L)

    char* ws = (char*)d_ws;
    const size_t wBytes  = (size_t)Lv * Dv * sizeof(float);          // 512 KB
    const size_t rsBytes = (size_t)Lv * Mv * sizeof(float);          // 128 KB
    size_t off = 0;
    float* W  = (float*)(ws + off); off += (wBytes  + 255) & ~(size_t)255;
    float* Rs = (float*)(ws + off); off += (rsBytes + 255) & ~(size_t)255;
    float* S  = (float*)(ws + off);                                   // 128 MB (B,L,N)

    k_normalize_rows<<<Lv, 256, 0, stream>>>(theta, W);
    k_project_wmma<<<dim3(Nv / 128, Lv / 16, Bv), 256, 0, stream>>>(X, W, S);
    k_ref_sorted<<<Lv, 128, 0, stream>>>(refp, W, Rs);
    k_sort_interp_emb<<<dim3(Lv, Bv), 256, 0, stream>>>(S, Rs, out);
}